// GraphSAGE_8830452760939
// MI455X (gfx1250) — compile-verified
//
#include <hip/hip_runtime.h>
#include <math.h>

#define N_NODES 100000
#define N_EDGES 1600000
#define D_IN 128
#define H 64
#define BN_EPS 1e-5f

typedef __attribute__((ext_vector_type(2))) float v2f;
typedef __attribute__((ext_vector_type(8))) float v8f;

// D = A(16x4 f32) * B(4x16 f32) + C(16x16 f32), full-precision WMMA on gfx1250
__device__ __forceinline__ v8f wmma4(v2f a, v2f b, v8f c) {
  return __builtin_amdgcn_wmma_f32_16x16x4_f32(false, a, false, b, (short)0, c,
                                               false, false);
}

// ---------------------------------------------------------------------------
// Embedding GEMM: out[M,64] = x[M,128] @ W^T + b       (W is [64,128] row-major)
// 1 wave = 16 rows x 64 cols. A tile staged in LDS (pad +4 dwords -> no bank
// conflicts on the b64 fragment reads). M = 6250 tiles exactly.
// ---------------------------------------------------------------------------
__global__ void k_emb(const float* __restrict__ x, const float* __restrict__ W,
                      const float* __restrict__ b, float* __restrict__ out) {
  __shared__ float lds[2][16 * (D_IN + 4)];
  const int lane = threadIdx.x & 31;
  const int wave = threadIdx.x >> 5;
  const int m0 = (blockIdx.x * 2 + wave) * 16;
  float* tl = lds[wave];

  // stage 16x128 A tile, coalesced float4 loads
  for (int idx = lane; idx < 16 * (D_IN / 4); idx += 32) {
    const int r = idx >> 5;          // 32 float4 per row
    const int c4 = idx & 31;
    const float4 v = *(const float4*)(x + (size_t)(m0 + r) * D_IN + c4 * 4);
    *(float4*)(tl + r * (D_IN + 4) + c4 * 4) = v;
  }
  __syncthreads();

  const int half = lane >> 4;
  const int l15 = lane & 15;
  v8f acc[4];
#pragma unroll
  for (int nt = 0; nt < 4; ++nt) {
    const float bv = b[nt * 16 + l15];
#pragma unroll
    for (int j = 0; j < 8; ++j) acc[nt][j] = bv;
  }

  for (int k0 = 0; k0 < D_IN; k0 += 4) {
    v2f a;
    const float* ap = tl + l15 * (D_IN + 4) + k0 + 2 * half;
    a.x = ap[0];
    a.y = ap[1];
#pragma unroll
    for (int nt = 0; nt < 4; ++nt) {
      const float* bp = W + (size_t)(nt * 16 + l15) * D_IN + k0 + 2 * half;
      v2f bb;
      bb.x = bp[0];
      bb.y = bp[1];
      acc[nt] = wmma4(a, bb, acc[nt]);
    }
  }

#pragma unroll
  for (int nt = 0; nt < 4; ++nt)
#pragma unroll
    for (int j = 0; j < 8; ++j)
      out[(size_t)(m0 + j + half * 8) * H + nt * 16 + l15] = acc[nt][j];
}

// ---------------------------------------------------------------------------
// SAGEConv GEMM: y[M,64] = (msg[M,64]*inv[m]) @ Wl^T + bl + hin[M,64] @ Wr^T
// Two K=64 phases into the same accumulators. y may alias msg (each wave
// rewrites only the 16 rows it already staged).
// ---------------------------------------------------------------------------
__global__ void k_conv(const float* msg, const float* __restrict__ inv,
                       const float* __restrict__ hin,
                       const float* __restrict__ Wl, const float* __restrict__ bl,
                       const float* __restrict__ Wr, float* y) {
  __shared__ float lds[2][16 * (H + 4)];
  const int lane = threadIdx.x & 31;
  const int wave = threadIdx.x >> 5;
  const int m0 = (blockIdx.x * 2 + wave) * 16;
  float* tl = lds[wave];
  const int half = lane >> 4;
  const int l15 = lane & 15;

  v8f acc[4];
#pragma unroll
  for (int nt = 0; nt < 4; ++nt) {
    const float bv = bl[nt * 16 + l15];
#pragma unroll
    for (int j = 0; j < 8; ++j) acc[nt][j] = bv;
  }

  // ---- phase A: aggregated neighbors (msg * inv) through Wl ----
  for (int idx = lane; idx < 16 * (H / 4); idx += 32) {
    const int r = idx >> 4;          // 16 float4 per row
    const int c4 = idx & 15;
    float4 v = *(const float4*)(msg + (size_t)(m0 + r) * H + c4 * 4);
    const float s = inv[m0 + r];
    v.x *= s; v.y *= s; v.z *= s; v.w *= s;
    *(float4*)(tl + r * (H + 4) + c4 * 4) = v;
  }
  __syncthreads();
  for (int k0 = 0; k0 < H; k0 += 4) {
    v2f a;
    const float* ap = tl + l15 * (H + 4) + k0 + 2 * half;
    a.x = ap[0];
    a.y = ap[1];
#pragma unroll
    for (int nt = 0; nt < 4; ++nt) {
      const float* bp = Wl + (size_t)(nt * 16 + l15) * H + k0 + 2 * half;
      v2f bb;
      bb.x = bp[0];
      bb.y = bp[1];
      acc[nt] = wmma4(a, bb, acc[nt]);
    }
  }
  __syncthreads();

  // ---- phase B: root features through Wr ----
  for (int idx = lane; idx < 16 * (H / 4); idx += 32) {
    const int r = idx >> 4;
    const int c4 = idx & 15;
    const float4 v = *(const float4*)(hin + (size_t)(m0 + r) * H + c4 * 4);
    *(float4*)(tl + r * (H + 4) + c4 * 4) = v;
  }
  __syncthreads();
  for (int k0 = 0; k0 < H; k0 += 4) {
    v2f a;
    const float* ap = tl + l15 * (H + 4) + k0 + 2 * half;
    a.x = ap[0];
    a.y = ap[1];
#pragma unroll
    for (int nt = 0; nt < 4; ++nt) {
      const float* bp = Wr + (size_t)(nt * 16 + l15) * H + k0 + 2 * half;
      v2f bb;
      bb.x = bp[0];
      bb.y = bp[1];
      acc[nt] = wmma4(a, bb, acc[nt]);
    }
  }

#pragma unroll
  for (int nt = 0; nt < 4; ++nt)
#pragma unroll
    for (int j = 0; j < 8; ++j)
      y[(size_t)(m0 + j + half * 8) * H + nt * 16 + l15] = acc[nt][j];
}

// ---------------------------------------------------------------------------
// scatter-add: one thread per (edge, 16-float chunk). L2-resident atomics.
// ---------------------------------------------------------------------------
__global__ void k_scatter(const float* __restrict__ h, const int* __restrict__ ei,
                          float* __restrict__ msg, float* __restrict__ cnt) {
  const int tid = blockIdx.x * blockDim.x + threadIdx.x;
  if (tid >= N_EDGES * 4) return;
  const int e = tid >> 2;
  const int c = tid & 3;
  const int src = ei[e];
  const int dst = ei[N_EDGES + e];
  const float* hp = h + (size_t)src * H + c * 16;
  float* mp = msg + (size_t)dst * H + c * 16;
#pragma unroll
  for (int i = 0; i < 4; ++i) {
    const float4 v = *(const float4*)(hp + i * 4);
    atomicAdd(mp + i * 4 + 0, v.x);
    atomicAdd(mp + i * 4 + 1, v.y);
    atomicAdd(mp + i * 4 + 2, v.z);
    atomicAdd(mp + i * 4 + 3, v.w);
  }
  if (c == 0) atomicAdd(cnt + dst, 1.0f);
}

__global__ void k_zero(float* __restrict__ p, int n) {
  const int i = blockIdx.x * blockDim.x + threadIdx.x;
  if (i < n) p[i] = 0.0f;
}

__global__ void k_inv(const float* __restrict__ cnt, float* __restrict__ inv) {
  const int i = blockIdx.x * blockDim.x + threadIdx.x;
  if (i < N_NODES) inv[i] = 1.0f / fmaxf(cnt[i], 1.0f);
}

// per-column sum / sumsq over 64 columns; coalesced 256B row reads
__global__ void k_bnstats(const float* __restrict__ y, float* __restrict__ sums) {
  const int c = threadIdx.x & 63;
  const int r0 = blockIdx.x * 256 + (threadIdx.x >> 6);
  float s = 0.0f, s2 = 0.0f;
  for (int i = 0; i < 64; ++i) {
    const int r = r0 + i * 4;
    if (r < N_NODES) {
      const float v = y[(size_t)r * H + c];
      s += v;
      s2 += v * v;
    }
  }
  atomicAdd(sums + c, s);
  atomicAdd(sums + 64 + c, s2);
}

__global__ void k_bnfin(const float* __restrict__ sums, const float* __restrict__ g,
                        const float* __restrict__ be, float* __restrict__ ss) {
  const int c = threadIdx.x;  // 64 threads
  const float mean = sums[c] * (1.0f / (float)N_NODES);
  const float var = sums[64 + c] * (1.0f / (float)N_NODES) - mean * mean;
  const float sc = g[c] * rsqrtf(var + BN_EPS);
  ss[c] = sc;
  ss[64 + c] = be[c] - mean * sc;
}

__global__ void k_bnrelu(const float* __restrict__ y, const float* __restrict__ ss,
                         float* __restrict__ h) {
  const int i = blockIdx.x * blockDim.x + threadIdx.x;
  if (i < N_NODES * H) {
    const int c = i & 63;
    h[i] = fmaxf(y[i] * ss[c] + ss[64 + c], 0.0f);
  }
}

// classifier (H=64 -> 2) + softmax, one thread per node
__global__ void k_cls(const float* __restrict__ h, const float* __restrict__ Wc,
                      const float* __restrict__ bc, float* __restrict__ out) {
  const int i = blockIdx.x * blockDim.x + threadIdx.x;
  if (i >= N_NODES) return;
  float a0 = bc[0], a1 = bc[1];
  const float* hp = h + (size_t)i * H;
#pragma unroll
  for (int k = 0; k < H; ++k) {
    const float v = hp[k];
    a0 += v * Wc[k];
    a1 += v * Wc[H + k];
  }
  const float m = fmaxf(a0, a1);
  const float e0 = expf(a0 - m);
  const float e1 = expf(a1 - m);
  const float r = 1.0f / (e0 + e1);
  out[2 * (size_t)i + 0] = e0 * r;
  out[2 * (size_t)i + 1] = e1 * r;
}

extern "C" void kernel_launch(void* const* d_in, const int* in_sizes, int n_in,
                              void* d_out, int out_size, void* d_ws, size_t ws_size,
                              hipStream_t stream) {
  (void)in_sizes; (void)n_in; (void)out_size; (void)ws_size;
  const float* x = (const float*)d_in[0];
  const int* ei = (const int*)d_in[1];
  const float* W_emb = (const float*)d_in[2];
  const float* b_emb = (const float*)d_in[3];
  const float* W_cls = (const float*)d_in[4];
  const float* b_cls = (const float*)d_in[5];
  float* out = (float*)d_out;

  // workspace layout (floats): h | msg(=y) | cnt | inv | bnsums(128) | scale+shift(128)
  float* h = (float*)d_ws;                        // N*H
  float* msg = h + (size_t)N_NODES * H;           // N*H, conv writes y in place
  float* cnt = msg + (size_t)N_NODES * H;         // N
  float* inv = cnt + N_NODES;                     // N
  float* sums = inv + N_NODES;                    // 128
  float* ss = sums + 128;                         // 128

  const dim3 gGemm(3125), bGemm(64);              // 6250 row tiles exactly
  k_emb<<<gGemm, bGemm, 0, stream>>>(x, W_emb, b_emb, h);

  for (int L = 0; L < 3; ++L) {
    const float* Wl = (const float*)d_in[6 + 5 * L];
    const float* bl = (const float*)d_in[7 + 5 * L];
    const float* Wr = (const float*)d_in[8 + 5 * L];
    const float* g = (const float*)d_in[9 + 5 * L];
    const float* be = (const float*)d_in[10 + 5 * L];

    k_zero<<<(N_NODES * H + 255) / 256, 256, 0, stream>>>(msg, N_NODES * H);
    k_zero<<<(N_NODES + 255) / 256, 256, 0, stream>>>(cnt, N_NODES);
    k_zero<<<1, 128, 0, stream>>>(sums, 128);
    k_scatter<<<(N_EDGES * 4 + 255) / 256, 256, 0, stream>>>(h, ei, msg, cnt);
    k_inv<<<(N_NODES + 255) / 256, 256, 0, stream>>>(cnt, inv);
    k_conv<<<gGemm, bGemm, 0, stream>>>(msg, inv, h, Wl, bl, Wr, msg);
    k_bnstats<<<(N_NODES + 255) / 256, 256, 0, stream>>>(msg, sums);
    k_bnfin<<<1, 64, 0, stream>>>(sums, g, be, ss);
    k_bnrelu<<<(N_NODES * H + 255) / 256, 256, 0, stream>>>(msg, ss, h);
  }

  k_cls<<<(N_NODES + 255) / 256, 256, 0, stream>>>(h, W_cls, b_cls, out);
}